// GNN_16140487098561
// MI455X (gfx1250) — compile-verified
//
#include <hip/hip_runtime.h>
#include <math.h>

typedef __attribute__((ext_vector_type(2))) float v2f;
typedef __attribute__((ext_vector_type(8))) float v8f;

// ---------------- degree / normalization ----------------
__global__ void k_deg_init(float* __restrict__ deg, int n) {
    int i = blockIdx.x * blockDim.x + threadIdx.x;
    if (i < n) deg[i] = 1.0f;  // self-loop weight
}

__global__ void k_deg_edges(float* __restrict__ deg, const int* __restrict__ dst,
                            const float* __restrict__ ew, int e) {
    int i = blockIdx.x * blockDim.x + threadIdx.x;
    if (i < e) atomicAdd(&deg[dst[i]], ew[i]);
}

__global__ void k_deg_rsqrt(float* __restrict__ deg, int n) {
    int i = blockIdx.x * blockDim.x + threadIdx.x;
    if (i < n) deg[i] = rsqrtf(deg[i]);  // deg >= 1 always (self-loop)
}

// Preload weights into LDS transposed + XOR-swizzled:
//   sW[c*K + (k ^ 4*(c&15))] = W[k*NC + c]
// -> B fragment (W[ka][c], W[ka+1][c]) is one contiguous 8B pair (ds_load_b64),
//    and banks are conflict-free across a half-wave for any even ka.
template <int K, int NC>
__device__ inline void preload_swizzled(const float* __restrict__ W, float* sW) {
    const float4* src = (const float4*)W;
    for (int i4 = threadIdx.x; i4 < K * NC / 4; i4 += blockDim.x) {
        float4 w = src[i4];
        int fi = i4 * 4;
        int k = fi / NC;
        int c = fi % NC;
        sW[(c + 0) * K + (k ^ (4 * ((c + 0) & 15)))] = w.x;
        sW[(c + 1) * K + (k ^ (4 * ((c + 1) & 15)))] = w.y;
        sW[(c + 2) * K + (k ^ (4 * ((c + 2) & 15)))] = w.z;
        sW[(c + 3) * K + (k ^ (4 * ((c + 3) & 15)))] = w.w;
    }
}

// ---------------- GEMM1: C[n x 64] = X[n x 256] @ W[256 x 64] ----------------
// WMMA f32 16x16x4. 8 waves / block; each wave owns a 16-row x 64-col strip.
__global__ void __launch_bounds__(256)
k_gemm1(const float* __restrict__ X, const float* __restrict__ W,
        float* __restrict__ C, int n) {
    __shared__ float sW[64 * 256];  // 64 KB, transposed + swizzled
    preload_swizzled<256, 64>(W, sW);
    __syncthreads();

    const int wv = threadIdx.x >> 5;
    const int lane = threadIdx.x & 31;
    const int row0 = blockIdx.x * 128 + wv * 16;
    if (row0 >= n) return;

    const int m = lane & 15;   // M index (A) / N index (B,D)
    const int hi = lane >> 4;  // selects K pair on A/B, M+8 on D
    const int xm = 4 * m;      // per-lane swizzle constant
    const float* xrow = X + (size_t)(row0 + m) * 256;

    v8f acc[4] = {};
    for (int k0 = 0; k0 < 256; k0 += 4) {
        const int ka = k0 + 2 * hi;
        v2f a = *(const v2f*)(xrow + ka);
        const int kx = ka ^ xm;
#pragma unroll
        for (int t = 0; t < 4; ++t) {
            v2f b = *(const v2f*)(sW + (t * 16 + m) * 256 + kx);
            acc[t] = __builtin_amdgcn_wmma_f32_16x16x4_f32(
                false, a, false, b, (short)0, acc[t], false, false);
        }
    }
#pragma unroll
    for (int t = 0; t < 4; ++t)
#pragma unroll
        for (int vr = 0; vr < 8; ++vr)
            C[(size_t)(row0 + vr + 8 * hi) * 64 + t * 16 + m] = acc[t][vr];
}

// ---------------- GEMM2: C[n x 128] = relu(H + b1)[n x 64] @ W[64 x 128] -----
__global__ void __launch_bounds__(256)
k_gemm2(const float* __restrict__ Hin, const float* __restrict__ bias1,
        const float* __restrict__ W, float* __restrict__ C, int n) {
    __shared__ float sW[128 * 64];  // 32 KB, transposed + swizzled
    preload_swizzled<64, 128>(W, sW);
    __syncthreads();

    const int wv = threadIdx.x >> 5;
    const int lane = threadIdx.x & 31;
    const int row0 = blockIdx.x * 128 + wv * 16;
    if (row0 >= n) return;

    const int m = lane & 15;
    const int hi = lane >> 4;
    const int xm = 4 * m;
    const float* hrow = Hin + (size_t)(row0 + m) * 64;

    v8f acc[8] = {};
    for (int k0 = 0; k0 < 64; k0 += 4) {
        const int ka = k0 + 2 * hi;
        v2f h = *(const v2f*)(hrow + ka);
        v2f bb = *(const v2f*)(bias1 + ka);
        v2f a;
        a.x = fmaxf(h.x + bb.x, 0.0f);  // fused b1 + ReLU
        a.y = fmaxf(h.y + bb.y, 0.0f);
        const int kx = ka ^ xm;
#pragma unroll
        for (int t = 0; t < 8; ++t) {
            v2f b = *(const v2f*)(sW + (t * 16 + m) * 64 + kx);
            acc[t] = __builtin_amdgcn_wmma_f32_16x16x4_f32(
                false, a, false, b, (short)0, acc[t], false, false);
        }
    }
#pragma unroll
    for (int t = 0; t < 8; ++t)
#pragma unroll
        for (int vr = 0; vr < 8; ++vr)
            C[(size_t)(row0 + vr + 8 * hi) * 128 + t * 16 + m] = acc[t][vr];
}

// ---------------- aggregation: self-loop init + edge scatter ----------------
template <int F>
__global__ void k_agg_init(const float* __restrict__ xw, const float* __restrict__ dis,
                           const float* __restrict__ bias, float* __restrict__ out,
                           int n) {
    long long i = (long long)blockIdx.x * blockDim.x + threadIdx.x;
    long long total = (long long)n * F;
    if (i >= total) return;
    int node = (int)(i / F);
    int f = (int)(i - (long long)node * F);
    float d = dis[node];
    float v = d * d * xw[i];  // self-loop: dis^2 * xw
    if (bias) v += bias[f];
    out[i] = v;
}

template <int F>
__global__ void k_agg_edges(const float* __restrict__ xw, const float* __restrict__ dis,
                            const int* __restrict__ src, const int* __restrict__ dst,
                            const float* __restrict__ ew, float* __restrict__ out,
                            int e) {
    int gid = blockIdx.x * blockDim.x + threadIdx.x;
    int edge = gid >> 5;
    int lane = gid & 31;
    if (edge >= e) return;
    int s = src[edge];
    int d = dst[edge];
    float c = dis[s] * ew[edge] * dis[d];
    const float* xs = xw + (size_t)s * F;
    float* od = out + (size_t)d * F;
#pragma unroll
    for (int f = lane; f < F; f += 32)
        atomicAdd(&od[f], c * xs[f]);
}

// ---------------- head: sigmoid(h2[join] @ Wl + bl) ----------------
__global__ void k_head(const float* __restrict__ h2, const int* __restrict__ jidx,
                       const float* __restrict__ Wl, const float* __restrict__ bl,
                       float* __restrict__ out, int B) {
    int gid = blockIdx.x * blockDim.x + threadIdx.x;
    int b = gid >> 5;
    int lane = gid & 31;
    if (b >= B) return;
    const float* h = h2 + (size_t)jidx[b] * 128;
    float p = 0.0f;
#pragma unroll
    for (int f = lane; f < 128; f += 32) p += h[f] * Wl[f];
#pragma unroll
    for (int off = 16; off; off >>= 1) p += __shfl_xor(p, off, 32);
    if (lane == 0) out[b] = 1.0f / (1.0f + expf(-(p + bl[0])));
}

extern "C" void kernel_launch(void* const* d_in, const int* in_sizes, int n_in,
                              void* d_out, int out_size, void* d_ws, size_t ws_size,
                              hipStream_t stream) {
    const float* x  = (const float*)d_in[0];
    const int*   ei = (const int*)d_in[1];
    const int*   ji = (const int*)d_in[2];
    const float* ew = (const float*)d_in[3];
    const float* W1 = (const float*)d_in[4];
    const float* b1 = (const float*)d_in[5];
    const float* W2 = (const float*)d_in[6];
    const float* b2 = (const float*)d_in[7];
    const float* Wl = (const float*)d_in[8];
    const float* bl = (const float*)d_in[9];

    const int FIN = 256, H1 = 64, H2 = 128;
    const int n = in_sizes[0] / FIN;  // 100000
    const int e = in_sizes[3];        // 1600000
    const int B = in_sizes[2];        // 4096
    const int* src = ei;
    const int* dst = ei + e;

    // workspace layout (floats): dis[n] | xw1[n*64] | ag1[n*64] | xw2[n*128]
    // out2[n*128] overlays (xw1|ag1), which are dead by the time it is written
    float* ws   = (float*)d_ws;
    float* dis  = ws;
    float* xw1  = ws + n;
    float* ag1  = xw1 + (size_t)n * H1;
    float* xw2  = ag1 + (size_t)n * H1;
    float* out2 = xw1;

    const dim3 blk(256);
    k_deg_init <<<(n + 255) / 256, blk, 0, stream>>>(dis, n);
    k_deg_edges<<<(e + 255) / 256, blk, 0, stream>>>(dis, dst, ew, e);
    k_deg_rsqrt<<<(n + 255) / 256, blk, 0, stream>>>(dis, n);

    k_gemm1<<<(n + 127) / 128, 256, 0, stream>>>(x, W1, xw1, n);

    {
        long long tot = (long long)n * H1;
        k_agg_init<64><<<(unsigned)((tot + 255) / 256), blk, 0, stream>>>(
            xw1, dis, nullptr, ag1, n);
    }
    k_agg_edges<64><<<(unsigned)(((long long)e * 32 + 255) / 256), blk, 0, stream>>>(
        xw1, dis, src, dst, ew, ag1, e);

    k_gemm2<<<(n + 127) / 128, 256, 0, stream>>>(ag1, b1, W2, xw2, n);

    {
        long long tot = (long long)n * H2;
        k_agg_init<128><<<(unsigned)((tot + 255) / 256), blk, 0, stream>>>(
            xw2, dis, b2, out2, n);
    }
    k_agg_edges<128><<<(unsigned)(((long long)e * 32 + 255) / 256), blk, 0, stream>>>(
        xw2, dis, src, dst, ew, out2, e);

    k_head<<<(B * 32 + 255) / 256, blk, 0, stream>>>(out2, ji, Wl, bl,
                                                     (float*)d_out, B);
}